// SplineConv_29205777613549
// MI455X (gfx1250) — compile-verified
//
#include <hip/hip_runtime.h>

#define NN   50000
#define EE   1600000
#define CIN  32
#define COUT 32
#define KK   25
#define KROW 800   // KK * CIN
#define KS5  5

typedef float v2f __attribute__((ext_vector_type(2)));
typedef float v8f __attribute__((ext_vector_type(8)));

// ---------------------------------------------------------------------------
// Zero a buffer (float4 granularity). Used for deg + M (and fallback buffers).
// ---------------------------------------------------------------------------
__global__ void spline_zero_f4(float4* __restrict__ p, int n4) {
    int i = blockIdx.x * blockDim.x + threadIdx.x;
    if (i < n4) p[i] = make_float4(0.f, 0.f, 0.f, 0.f);
}

// ---------------------------------------------------------------------------
// Per-edge basis helper (degree-1 open B-spline, KS=5, DIM=2).
// All lanes compute redundantly (scalar-cheap).
// ---------------------------------------------------------------------------
__device__ __forceinline__ void spline_basis(float p0, float p1,
                                             float bas[4], int wi[4]) {
    float v0 = p0 * 4.0f, v1 = p1 * 4.0f;           // pseudo * (KS - 1)
    float f0 = floorf(v0), f1 = floorf(v1);
    float fr0 = v0 - f0,   fr1 = v1 - f1;
    int   i0 = (int)f0,    i1 = (int)f1;
#pragma unroll
    for (int s = 0; s < 4; ++s) {
        int o0 = s & 1, o1 = (s >> 1) & 1;
        float w0 = o0 ? fr0 : 1.0f - fr0;
        float w1 = o1 ? fr1 : 1.0f - fr1;
        bas[s] = w0 * w1;
        int id0 = i0 + o0; if (id0 >= KS5) id0 -= KS5;
        int id1 = i1 + o1; if (id1 >= KS5) id1 -= KS5;
        wi[s] = id0 + KS5 * id1;                     // stride = [1, 5]
    }
}

// ---------------------------------------------------------------------------
// Phase A: edge scatter into M[N, K, IN] (L2-resident, 160 MB) + degree.
// One wave per edge; lane = input channel -> each atomic burst is 128B
// contiguous (cacheline-coalesced L2 atomics).
// ---------------------------------------------------------------------------
__global__ void spline_scatter(const float* __restrict__ x,
                               const int*   __restrict__ ei,
                               const float* __restrict__ pseudo,
                               float* __restrict__ M,
                               float* __restrict__ deg) {
    const int lane = threadIdx.x;                    // 0..31, channel
    const int e    = blockIdx.x * blockDim.y + threadIdx.y;
    if (e >= EE) return;

    const int row = ei[e];
    const int col = ei[EE + e];
    float bas[4]; int wi[4];
    spline_basis(pseudo[2 * e], pseudo[2 * e + 1], bas, wi);

    const float xv = x[(size_t)col * CIN + lane];
    if (lane == 0) unsafeAtomicAdd(&deg[row], 1.0f);

#pragma unroll
    for (int s = 0; s < 4; ++s) {
        float* dst = M + ((size_t)row * KK + wi[s]) * CIN + lane;
        unsafeAtomicAdd(dst, xv * bas[s]);
    }
}

// ---------------------------------------------------------------------------
// Phase B: out = (M[N,800] @ W[800,32]) / max(deg,1) + x @ root + bias
// One wave per 16-node tile, producing a 16x32 output tile with
// V_WMMA_F32_16X16X4_F32. Layouts per ISA 7.12.2:
//   A (16x4 f32):  M = lane%16, K = vgpr + 2*(lane/16)
//   B (4x16 f32):  K = vgpr + 2*(lane/16), N = lane%16
//   C/D (16x16):   M = vgpr + 8*(lane/16), N = lane%16
// ---------------------------------------------------------------------------
__global__ void spline_gemm(const float* __restrict__ M,
                            const float* __restrict__ W,
                            const float* __restrict__ deg,
                            const float* __restrict__ x,
                            const float* __restrict__ root,
                            const float* __restrict__ bias,
                            float* __restrict__ out) {
    const int lane   = threadIdx.x & 31;
    const int wave   = (blockIdx.x * blockDim.x + threadIdx.x) >> 5;
    const int tiles  = NN / 16;                      // 3125
    if (wave >= tiles) return;                       // wave-uniform exit

    const int base   = wave * 16;
    const int laneLo = lane & 15;
    const int laneHi = lane >> 4;

    v8f acc0 = {};
    v8f acc1 = {};

    const float* Arow = M + (size_t)(base + laneLo) * KROW + 2 * laneHi;

    // ---- main contraction over K*IN = 800 (200 k-blocks of 4) ----
    for (int kb = 0; kb < KROW / 4; ++kb) {
        v2f a = *(const v2f*)(Arow + kb * 4);        // K = kb*4 + 2*laneHi (+0,+1)
        const int kA = kb * 4 + 2 * laneHi;
        v2f b0, b1;
        b0.x = W[(size_t)kA * COUT + laneLo];
        b0.y = W[(size_t)(kA + 1) * COUT + laneLo];
        b1.x = W[(size_t)kA * COUT + 16 + laneLo];
        b1.y = W[(size_t)(kA + 1) * COUT + 16 + laneLo];
        acc0 = __builtin_amdgcn_wmma_f32_16x16x4_f32(false, a, false, b0,
                                                     (short)0, acc0, false, false);
        acc1 = __builtin_amdgcn_wmma_f32_16x16x4_f32(false, a, false, b1,
                                                     (short)0, acc1, false, false);
    }

    // ---- degree normalization (applies only to the spline term) ----
#pragma unroll
    for (int r = 0; r < 8; ++r) {
        float d = deg[base + r + 8 * laneHi];
        float s = 1.0f / fmaxf(d, 1.0f);
        acc0[r] *= s;
        acc1[r] *= s;
    }

    // ---- fold in x @ root as 8 more WMMA k-blocks ----
    const float* Xrow = x + (size_t)(base + laneLo) * CIN + 2 * laneHi;
#pragma unroll
    for (int kb = 0; kb < CIN / 4; ++kb) {
        v2f a = *(const v2f*)(Xrow + kb * 4);
        const int kA = kb * 4 + 2 * laneHi;
        v2f b0, b1;
        b0.x = root[(size_t)kA * COUT + laneLo];
        b0.y = root[(size_t)(kA + 1) * COUT + laneLo];
        b1.x = root[(size_t)kA * COUT + 16 + laneLo];
        b1.y = root[(size_t)(kA + 1) * COUT + 16 + laneLo];
        acc0 = __builtin_amdgcn_wmma_f32_16x16x4_f32(false, a, false, b0,
                                                     (short)0, acc0, false, false);
        acc1 = __builtin_amdgcn_wmma_f32_16x16x4_f32(false, a, false, b1,
                                                     (short)0, acc1, false, false);
    }

    // ---- bias + store (64B contiguous per half-wave per vgpr) ----
    const float bia0 = bias[laneLo];
    const float bia1 = bias[16 + laneLo];
#pragma unroll
    for (int r = 0; r < 8; ++r) {
        const size_t rowN = (size_t)(base + r + 8 * laneHi);
        out[rowN * COUT + laneLo]      = acc0[r] + bia0;
        out[rowN * COUT + 16 + laneLo] = acc1[r] + bia1;
    }
}

// ---------------------------------------------------------------------------
// Fallback path (workspace too small for M): per-edge direct transform.
// Wave per edge, lane = output channel; x row broadcast via __shfl.
// Accumulates into d_out, finalized by spline_finalize.
// ---------------------------------------------------------------------------
__global__ void spline_edge_direct(const float* __restrict__ x,
                                   const int*   __restrict__ ei,
                                   const float* __restrict__ pseudo,
                                   const float* __restrict__ W,
                                   float* __restrict__ outacc,
                                   float* __restrict__ deg) {
    const int lane = threadIdx.x;                    // output channel
    const int e    = blockIdx.x * blockDim.y + threadIdx.y;
    if (e >= EE) return;

    const int row = ei[e];
    const int col = ei[EE + e];
    float bas[4]; int wi[4];
    spline_basis(pseudo[2 * e], pseudo[2 * e + 1], bas, wi);

    const float xv = x[(size_t)col * CIN + lane];
    if (lane == 0) unsafeAtomicAdd(&deg[row], 1.0f);

    float acc = 0.0f;
#pragma unroll 4
    for (int i = 0; i < CIN; ++i) {
        float xb = __shfl(xv, i, 32);
        float ws = 0.0f;
#pragma unroll
        for (int s = 0; s < 4; ++s)
            ws += bas[s] * W[((size_t)wi[s] * CIN + i) * COUT + lane];
        acc += xb * ws;
    }
    unsafeAtomicAdd(&outacc[(size_t)row * COUT + lane], acc);
}

__global__ void spline_finalize(float* __restrict__ out,
                                const float* __restrict__ deg,
                                const float* __restrict__ x,
                                const float* __restrict__ root,
                                const float* __restrict__ bias) {
    int idx = blockIdx.x * blockDim.x + threadIdx.x;
    if (idx >= NN * COUT) return;
    int n = idx >> 5, o = idx & 31;
    float v = out[idx] / fmaxf(deg[n], 1.0f);
    float r = 0.0f;
#pragma unroll
    for (int i = 0; i < CIN; ++i)
        r += x[(size_t)n * CIN + i] * root[(size_t)i * COUT + o];
    out[idx] = v + r + bias[o];
}

// ---------------------------------------------------------------------------
extern "C" void kernel_launch(void* const* d_in, const int* in_sizes, int n_in,
                              void* d_out, int out_size, void* d_ws, size_t ws_size,
                              hipStream_t stream) {
    const float* x      = (const float*)d_in[0];
    const int*   ei     = (const int*)  d_in[1];
    const float* pseudo = (const float*)d_in[2];
    const float* weight = (const float*)d_in[3];
    const float* root   = (const float*)d_in[4];
    const float* bias   = (const float*)d_in[5];
    float*       out    = (float*)d_out;

    const size_t needM = ((size_t)NN + (size_t)NN * KK * CIN) * sizeof(float);

    if (ws_size >= needM) {
        // ---- main path: scatter into M, then WMMA GEMM ----
        float* deg = (float*)d_ws;
        float* M   = deg + NN;

        const int ztot = NN + NN * KK * CIN;         // 40,050,000 (divisible by 4)
        const int n4   = ztot / 4;
        spline_zero_f4<<<(n4 + 255) / 256, 256, 0, stream>>>((float4*)d_ws, n4);

        dim3 blk(32, 8);
        spline_scatter<<<(EE + 7) / 8, blk, 0, stream>>>(x, ei, pseudo, M, deg);

        const int tiles = NN / 16;                   // 3125 waves
        const int thr   = tiles * 32;
        spline_gemm<<<(thr + 255) / 256, 256, 0, stream>>>(M, weight, deg, x,
                                                           root, bias, out);
    } else {
        // ---- fallback: per-edge direct transform (needs only deg in ws) ----
        float* deg = (float*)d_ws;
        spline_zero_f4<<<((NN / 4) + 255) / 256, 256, 0, stream>>>((float4*)deg, NN / 4);
        spline_zero_f4<<<((NN * COUT / 4) + 255) / 256, 256, 0, stream>>>(
            (float4*)out, NN * COUT / 4);

        dim3 blk(32, 8);
        spline_edge_direct<<<(EE + 7) / 8, blk, 0, stream>>>(x, ei, pseudo,
                                                             weight, out, deg);
        spline_finalize<<<(NN * COUT + 255) / 256, 256, 0, stream>>>(out, deg, x,
                                                                     root, bias);
    }
}